// Cell12AccModule_9732395893065
// MI455X (gfx1250) — compile-verified
//
#include <hip/hip_runtime.h>
#include <stdint.h>

#define C      16
#define NCOLS  49          // 3*C + 1
#define TILE   256         // pixels per tile == blockDim.x of scatter kernel

// ---- order-preserving float <-> uint32 map (exact, deterministic min/max) ----
__device__ __forceinline__ uint32_t f2mono(float f) {
  uint32_t b = __float_as_uint(f);
  return (b & 0x80000000u) ? ~b : (b | 0x80000000u);
}
__device__ __forceinline__ float mono2f(uint32_t u) {
  return __uint_as_float((u & 0x80000000u) ? (u ^ 0x80000000u) : ~u);
}

// -------------------- init: min cols = +inf(encoded), rest = 0 ---------------
__global__ void __launch_bounds__(256)
cell12_init_kernel(uint32_t* __restrict__ out, unsigned total) {
  unsigned i = blockIdx.x * 256u + threadIdx.x;
  if (i >= total) return;
  unsigned col = i % (unsigned)NCOLS;            // 32-bit magic-mul, cheap
  out[i] = (col < C) ? 0xFFFFFFFFu : 0u;         // umax for min-accum; 0 for max/sum
}

// -------------------- per-pixel accumulation (96 L2 atomics) -----------------
__device__ __forceinline__ void accum_pixel(const float* __restrict__ vals,
                                            uint32_t* __restrict__ o1u,
                                            float*    __restrict__ o1f, long b1,
                                            uint32_t* __restrict__ o2u,
                                            float*    __restrict__ o2f, long b2) {
  uint32_t* __restrict__ p1 = o1u + b1;
  float*    __restrict__ q1 = o1f + b1;
  uint32_t* __restrict__ p2 = o2u + b2;
  float*    __restrict__ q2 = o2f + b2;
#pragma unroll
  for (int c = 0; c < C; ++c) {
    float val  = vals[c];
    uint32_t e = f2mono(val);
    __hip_atomic_fetch_min(p1 + c,         e,   __ATOMIC_RELAXED, __HIP_MEMORY_SCOPE_AGENT);
    __hip_atomic_fetch_max(p1 + C + c,     e,   __ATOMIC_RELAXED, __HIP_MEMORY_SCOPE_AGENT);
    __hip_atomic_fetch_add(q1 + 2*C + c,   val, __ATOMIC_RELAXED, __HIP_MEMORY_SCOPE_AGENT);
    __hip_atomic_fetch_min(p2 + c,         e,   __ATOMIC_RELAXED, __HIP_MEMORY_SCOPE_AGENT);
    __hip_atomic_fetch_max(p2 + C + c,     e,   __ATOMIC_RELAXED, __HIP_MEMORY_SCOPE_AGENT);
    __hip_atomic_fetch_add(q2 + 2*C + c,   val, __ATOMIC_RELAXED, __HIP_MEMORY_SCOPE_AGENT);
  }
}

// --------- scatter: async-LDS double-buffered pipeline over pixel tiles ------
__global__ void __launch_bounds__(TILE)
cell12_scatter_kernel(const float* __restrict__ x,
                      const int*   __restrict__ m1,
                      const int*   __restrict__ m2,
                      uint32_t* __restrict__ o1u, float* __restrict__ o1f,
                      uint32_t* __restrict__ o2u, float* __restrict__ o2f,
                      int ntiles, long npix) {
  __shared__ __align__(16) float smem[2][TILE * C];   // 2 x 16 KB
  const int tid    = threadIdx.x;
  const int stride = gridDim.x;

  const uint32_t lds[2] = {(uint32_t)(uintptr_t)&smem[0][tid * C],
                           (uint32_t)(uintptr_t)&smem[1][tid * C]};

  int tile = blockIdx.x;

  // prologue: async-stage first tile into buffer 0 (each lane moves its 64B row)
  if (tile < ntiles) {
    uint32_t goff = ((uint32_t)tile * TILE + (uint32_t)tid) * (C * 4);
#pragma unroll
    for (int j = 0; j < 4; ++j)
      asm volatile("global_load_async_to_lds_b128 %0, %1, %2 offset:0 th:TH_LOAD_NT"
                   :: "v"(lds[0] + 16u * j), "v"(goff + 16u * j), "s"((const void*)x)
                   : "memory");
  }

  int buf = 0;
  for (; tile < ntiles; tile += stride, buf ^= 1) {
    // Issue label loads FIRST so their latency overlaps the async wait below.
    const long p   = (long)tile * TILE + tid;
    const long b1  = (long)__builtin_nontemporal_load(m1 + p) * NCOLS;
    const long b2  = (long)__builtin_nontemporal_load(m2 + p) * NCOLS;

    const int nxt = tile + stride;
    if (nxt < ntiles) {
      // prefetch next tile into the other buffer, then wait for current tile
      uint32_t goff = ((uint32_t)nxt * TILE + (uint32_t)tid) * (C * 4);
      const uint32_t dst = lds[buf ^ 1];
#pragma unroll
      for (int j = 0; j < 4; ++j)
        asm volatile("global_load_async_to_lds_b128 %0, %1, %2 offset:0 th:TH_LOAD_NT"
                     :: "v"(dst + 16u * j), "v"(goff + 16u * j), "s"((const void*)x)
                     : "memory");
      asm volatile("s_wait_asynccnt 4" ::: "memory");   // oldest 4 (current buf) done
    } else {
      asm volatile("s_wait_asynccnt 0" ::: "memory");
    }

    float vals[C];
    const float4* rp = (const float4*)&smem[buf][tid * C];
    *(float4*)&vals[0]  = rp[0];
    *(float4*)&vals[4]  = rp[1];
    *(float4*)&vals[8]  = rp[2];
    *(float4*)&vals[12] = rp[3];

    accum_pixel(vals, o1u, o1f, b1, o2u, o2f, b2);
  }

  // tail pixels (npix not a multiple of TILE) — plain global loads
  for (long p = (long)ntiles * TILE + (long)blockIdx.x * TILE + tid; p < npix;
       p += (long)stride * TILE) {
    float vals[C];
    const float4* xr = (const float4*)(x + p * C);
    *(float4*)&vals[0]  = xr[0];
    *(float4*)&vals[4]  = xr[1];
    *(float4*)&vals[8]  = xr[2];
    *(float4*)&vals[12] = xr[3];
    accum_pixel(vals, o1u, o1f, (long)m1[p] * NCOLS, o2u, o2f, (long)m2[p] * NCOLS);
  }
}

// --------- finalize: decode min/max, sum->mean, size feature column ----------
__global__ void __launch_bounds__(256)
cell12_finalize_kernel(uint32_t* __restrict__ ou,
                       float*    __restrict__ of,
                       const int* __restrict__ sizes,
                       unsigned total) {
  unsigned i = blockIdx.x * 256u + threadIdx.x;
  if (i >= total) return;
  unsigned row = i / (unsigned)NCOLS;            // 32-bit magic-mul
  unsigned col = i - row * (unsigned)NCOLS;
  if (col < 2 * C) {
    of[i] = mono2f(ou[i]);                       // decode min / max
  } else if (col < 3 * C) {
    of[i] = of[i] / (float)sizes[row];           // sum -> mean
  } else {
    of[i] = expf(-(float)sizes[row]) - 0.5f;     // size feature
  }
}

extern "C" void kernel_launch(void* const* d_in, const int* in_sizes, int n_in,
                              void* d_out, int out_size, void* d_ws, size_t ws_size,
                              hipStream_t stream) {
  const float* x  = (const float*)d_in[0];
  const int*   m1 = (const int*)d_in[1];
  const int*   m2 = (const int*)d_in[2];
  const int*   s1 = (const int*)d_in[3];
  const int*   s2 = (const int*)d_in[4];

  const long npix = in_sizes[1];        // N (length of cell_1_mask)
  const long L1   = in_sizes[3];        // length of cell_1_sizes
  const long L2   = in_sizes[4];        // length of cell_2_sizes

  float* out1 = (float*)d_out;
  float* out2 = out1 + L1 * NCOLS;

  // 1) init accumulators across the whole d_out
  {
    unsigned tot = (unsigned)((L1 + L2) * NCOLS);
    cell12_init_kernel<<<(tot + 255u) / 256u, 256, 0, stream>>>((uint32_t*)d_out, tot);
  }

  // 2) pipelined scatter
  {
    int ntiles = (int)(npix / TILE);
    int blocks = 2048;
    if (ntiles > 0 && blocks > ntiles) blocks = ntiles;
    if (ntiles == 0) blocks = 1;        // tail-only degenerate case
    cell12_scatter_kernel<<<blocks, TILE, 0, stream>>>(
        x, m1, m2,
        (uint32_t*)out1, out1,
        (uint32_t*)out2, out2,
        ntiles, npix);
  }

  // 3) finalize both outputs
  {
    unsigned t1 = (unsigned)(L1 * NCOLS);
    cell12_finalize_kernel<<<(t1 + 255u) / 256u, 256, 0, stream>>>(
        (uint32_t*)out1, out1, s1, t1);
    unsigned t2 = (unsigned)(L2 * NCOLS);
    cell12_finalize_kernel<<<(t2 + 255u) / 256u, 256, 0, stream>>>(
        (uint32_t*)out2, out2, s2, t2);
  }
}